// KVGather_6073083757123
// MI455X (gfx1250) — compile-verified
//
#include <hip/hip_runtime.h>
#include <hip/hip_bf16.h>

// KV gather-scale: out[b,i,k,:,:] = r_weight[b,i,k] * kv[b, r_idx[b,i,k], :, :]
// Pure bandwidth problem (AI ~ 0.125 FLOP/byte) -> no WMMA applies. MI455X
// strategy:
//   - one block per (b,i,k) tile (6272 blocks, 256 threads = 8 wave32)
//   - gathered kv slab staged via CDNA5 async global->LDS (ASYNCcnt) with
//     double buffering, 16 chunks x 4KB, s_wait_asynccnt pipelining
//   - explicit s_wait_dscnt guard before recycling an LDS buffer (async LDS
//     writes are unordered vs DScnt ops -> avoid WAR on the double buffer)
//   - 128-bit non-temporal stores so the 411MB write-once stream does not
//     evict the 103MB kv working set from the 192MB L2 (kv re-read topk=4x)

typedef float __attribute__((ext_vector_type(4))) vf4;

#define N_     32
#define P2_    49
#define TOPK_  4
#define W2_    64
#define CKV_   256
#define TILE_FLOATS (W2_ * CKV_)           // 16384 floats = 64 KB per tile
#define CHUNK  256                         // float4 per chunk == blockDim.x
#define NCHUNK (TILE_FLOATS / 4 / CHUNK)   // 16 chunks per tile

// Raw LDS byte offset (what VDST of global_load_async_to_lds_* expects):
// generic -> addrspace(3) cast, then ptrtoint.
__device__ __forceinline__ unsigned lds_byte_off(const void* p) {
  return (unsigned)(unsigned long long)(__attribute__((address_space(3))) const void*)p;
}

// One b128 async copy per lane: 16 bytes from global address g to LDS offset lds_off (GV mode, 64-bit
// vaddr). Tracked by ASYNCcnt; completion enforced with s_wait_asynccnt.
__device__ __forceinline__ void async_copy_b128(unsigned lds_off, const vf4* g) {
  asm volatile("global_load_async_to_lds_b128 %0, %1, off"
               :
               : "v"(lds_off), "v"((unsigned long long)(uintptr_t)g)
               : "memory");
}

__global__ __launch_bounds__(CHUNK)
void KVGather_6073083757123_kernel(const int*   __restrict__ r_idx,
                                   const float* __restrict__ r_weight,
                                   const float* __restrict__ kv,
                                   float*       __restrict__ out) {
  __shared__ vf4 sbuf[2][CHUNK];   // 8 KB double buffer

  const int bid = blockIdx.x;      // flat (b, i, k); matches r_idx/r_weight layout
  const int tid = threadIdx.x;
  const int b   = bid / (P2_ * TOPK_);

  int j = r_idx[bid];              // gather index into p2 axis (uniform -> s_load)
  j = j < 0 ? 0 : (j >= P2_ ? P2_ - 1 : j);
  const float w = r_weight[bid];   // uniform scalar weight

  const vf4* __restrict__ src =
      (const vf4*)(kv + ((size_t)b * P2_ + (size_t)j) * (size_t)TILE_FLOATS);
  vf4* __restrict__ dst = (vf4*)(out + (size_t)bid * (size_t)TILE_FLOATS);

  const unsigned l0 = lds_byte_off(&sbuf[0][tid]);
  const unsigned l1 = lds_byte_off(&sbuf[1][tid]);

  // Prologue: chunk 0 in flight.
  async_copy_b128(l0, src + tid);

#pragma unroll
  for (int c = 0; c < NCHUNK; ++c) {
    if (c + 1 < NCHUNK) {
      // The buffer we are about to refill was ds_load-ed in iteration c-1;
      // async LDS writes are unordered vs DScnt, so retire pending LDS reads
      // before recycling it (WAR guard; ~free, the read was already consumed).
      asm volatile("s_wait_dscnt 0x0" ::: "memory");
      // Issue next chunk into the other buffer, then wait only for the
      // current one (async loads complete in order -> asynccnt <= 1).
      async_copy_b128(((c + 1) & 1) ? l1 : l0, src + (c + 1) * CHUNK + tid);
      asm volatile("s_wait_asynccnt 0x1" ::: "memory");
    } else {
      asm volatile("s_wait_asynccnt 0x0" ::: "memory");
    }
    // Each lane reads exactly the LDS slot its own async op filled: no
    // cross-wave sharing, so no workgroup barrier is needed.
    vf4 v = sbuf[c & 1][tid];
    v *= w;
    // NT store: stream 411MB of write-once output around L2 so kv stays hot.
    __builtin_nontemporal_store(v, dst + c * CHUNK + tid);
  }
}

extern "C" void kernel_launch(void* const* d_in, const int* in_sizes, int n_in,
                              void* d_out, int out_size, void* d_ws, size_t ws_size,
                              hipStream_t stream) {
  (void)n_in; (void)out_size; (void)d_ws; (void)ws_size;
  const int*   r_idx    = (const int*)d_in[0];
  const float* r_weight = (const float*)d_in[1];
  const float* kv       = (const float*)d_in[2];
  float*       out      = (float*)d_out;

  const int nblocks = in_sizes[0];  // n * p2 * topk = 6272 tiles
  KVGather_6073083757123_kernel<<<nblocks, CHUNK, 0, stream>>>(r_idx, r_weight, kv, out);
}